// Classifier_57131654971429
// MI455X (gfx1250) — compile-verified
//
#include <hip/hip_runtime.h>
#include <hip/hip_bf16.h>

// MI455X / gfx1250, wave32. Fused 2-layer classifier head:
//  - fp32 WMMA (V_WMMA_F32_16X16X4_F32) for exact fp32 math
//  - double-buffered GLOBAL_LOAD_ASYNC_TO_LDS_B128 staging (ASYNCcnt pipeline)
//  - NT cache policy on the single-use x stream; RT for the reused W1

typedef __attribute__((ext_vector_type(2))) float v2f;
typedef __attribute__((ext_vector_type(8))) float v8f;
typedef __attribute__((ext_vector_type(4))) int   v4i;

#define FEAT 3136
#define HIDDEN 64
#define NCLS 21
#define SEL 128
#define RPB 64            // rows (proposals) per block
#define KC 32             // K-chunk for layer-1 staging
#define NCHUNK (FEAT / KC)
#define XS_STRIDE 36      // 32 + 4 pad  (bank-conflict-free A reads, 16B-alignable)
#define WS_STRIDE 72      // 64 + 8 pad  (disjoint bank sets for the two half-waves)
#define HS_STRIDE 68      // 64 + 4 pad
#define W2S_STRIDE 40     // 32 + 8 pad

#define CPOL_RT 0         // regular temporal (default)
#define CPOL_NT 1         // TH[2:0]=1: non-temporal (single-use stream)

#if __has_builtin(__builtin_amdgcn_global_load_async_to_lds_b128)
#define HAVE_ASYNC_LDS 1
#else
#define HAVE_ASYNC_LDS 0
#endif

#if HAVE_ASYNC_LDS
typedef __attribute__((address_space(1))) v4i gv4i_t;   // global v4i
typedef __attribute__((address_space(3))) v4i lv4i_t;   // LDS v4i
#define ASYNC_B128(g, l, cpol)                                                \
    __builtin_amdgcn_global_load_async_to_lds_b128(                           \
        (gv4i_t*)(void*)const_cast<float*>(g), (lv4i_t*)(void*)(l), 0, (cpol))
#if __has_builtin(__builtin_amdgcn_s_wait_asynccnt)
#define WAIT_ASYNC(n) __builtin_amdgcn_s_wait_asynccnt(n)
#else
#define WAIT_ASYNC(n) asm volatile("s_wait_asynccnt %0" ::"i"(n) : "memory")
#endif
#endif

__launch_bounds__(128)
__global__ void classifier_head_kernel(const float* __restrict__ x,
                                       const float* __restrict__ W1,
                                       const float* __restrict__ b1,
                                       const float* __restrict__ W2,
                                       const float* __restrict__ b2,
                                       const int*   __restrict__ keep_count,
                                       float* __restrict__ out) {
#if HAVE_ASYNC_LDS
    __shared__ float Xs[2][RPB * XS_STRIDE];     // 2 x  9216 B
    __shared__ float W1s[2][KC * WS_STRIDE];     // 2 x  9216 B
#else
    __shared__ float Xs[1][RPB * XS_STRIDE];
    __shared__ float W1s[1][KC * WS_STRIDE];
#endif
    __shared__ float Hs[RPB * HS_STRIDE];        // 17408 B
    __shared__ float W2s[HIDDEN * W2S_STRIDE];   // 10240 B

    const int tid  = threadIdx.x;
    const int wave = tid >> 5;                   // 0..3, one 16-row stripe each
    const int lane = tid & 31;
    const int rl   = lane & 15;                  // N-index / M-index within fragment
    const int kp   = (lane >> 4) << 1;           // K sub-offset: 0 or 2 (A/B layout)
    const int hi8  = (lane < 16) ? 0 : 8;        // C/D layout row offset
    const int wr0  = wave << 4;                  // wave's local row base

    const int b    = blockIdx.x >> 1;            // batch
    const int row0 = (blockIdx.x & 1) << 6;      // 0 or 64: which half of the 128 rows

    const float* xg = x + ((size_t)b * 300 + row0) * FEAT;

    // Per-thread staging coordinates (same mapping for sync and async paths).
    const int xr0 = tid >> 3,  xc0 = (tid & 7) << 2;    // X tile: 8 float4 per row
    const int wk0 = tid >> 4,  wc0 = (tid & 15) << 2;   // W1 tile: 16 float4 per row

#if HAVE_ASYNC_LDS
    // Issue one chunk's staging: 8 async b128 instructions per wave.
    auto stage_async = [&](int k0, int buf) {
#pragma unroll
        for (int j = 0; j < 4; ++j) {            // X tile: 64 rows x 32 k (stream, NT)
            int r = xr0 + j * 16;
            ASYNC_B128(xg + (size_t)r * FEAT + k0 + xc0,
                       &Xs[buf][r * XS_STRIDE + xc0], CPOL_NT);
        }
#pragma unroll
        for (int j = 0; j < 4; ++j) {            // W1 tile: 32 k x 64 n (L2-resident, RT)
            int kk = wk0 + j * 8;
            ASYNC_B128(W1 + (size_t)(k0 + kk) * HIDDEN + wc0,
                       &W1s[buf][kk * WS_STRIDE + wc0], CPOL_RT);
        }
    };
#else
    auto stage_sync = [&](int k0, int buf) {
#pragma unroll
        for (int j = 0; j < 4; ++j) {
            int r = xr0 + j * 16;
            float4 v = *(const float4*)(xg + (size_t)r * FEAT + k0 + xc0);
            *(float4*)(&Xs[buf][r * XS_STRIDE + xc0]) = v;
        }
#pragma unroll
        for (int j = 0; j < 4; ++j) {
            int kk = wk0 + j * 8;
            float4 v = *(const float4*)(W1 + (size_t)(k0 + kk) * HIDDEN + wc0);
            *(float4*)(&W1s[buf][kk * WS_STRIDE + wc0]) = v;
        }
    };
#endif

    // Stage zero-padded W2 (64 x 21 -> 64 x 32) once.
    for (int e = tid; e < HIDDEN * 32; e += 128) {
        int k = e >> 5, c = e & 31;
        W2s[k * W2S_STRIDE + c] = (c < NCLS) ? W2[k * NCLS + c] : 0.0f;
    }

    // Per-lane layer-1 bias values (col = t*16 + rl for each N-tile t).
    float bias1[4];
#pragma unroll
    for (int t = 0; t < 4; ++t) bias1[t] = b1[t * 16 + rl];

    // ---------------- Layer 1: H = leaky(X @ W1 + b1), K = 3136 ----------------
    v8f acc[4] = {};                             // 4 N-tiles of 16x16 fp32

#if HAVE_ASYNC_LDS
    stage_async(0, 0);                           // prefetch chunk 0
    for (int c = 0; c < NCHUNK; ++c) {
        const int buf = c & 1;
        if (c + 1 < NCHUNK) {
            stage_async((c + 1) * KC, buf ^ 1);  // prefetch next chunk
            WAIT_ASYNC(8);                       // chunk c's 8 loads done (in order)
        } else {
            WAIT_ASYNC(0);
        }
        __syncthreads();                         // all waves' chunk-c data in LDS
#else
    for (int c = 0; c < NCHUNK; ++c) {
        const int buf = 0;
        __syncthreads();                         // LDS WAR vs previous chunk
        stage_sync(c * KC, buf);
        __syncthreads();
#endif

#pragma unroll
        for (int kk = 0; kk < KC; kk += 4) {
            v2f a;                               // A frag: 16x4, row = rl, K = kk+kp..+1
            const float* ap = &Xs[buf][(wr0 + rl) * XS_STRIDE + kk + kp];
            a.x = ap[0];
            a.y = ap[1];
#pragma unroll
            for (int t = 0; t < 4; ++t) {        // B frag: 4x16, col = t*16 + rl
                v2f bb;
                bb.x = W1s[buf][(kk + kp)     * WS_STRIDE + t * 16 + rl];
                bb.y = W1s[buf][(kk + kp + 1) * WS_STRIDE + t * 16 + rl];
                acc[t] = __builtin_amdgcn_wmma_f32_16x16x4_f32(
                    false, a, false, bb, (short)0, acc[t], false, false);
            }
        }
#if HAVE_ASYNC_LDS
        __syncthreads();  // all waves done reading buf before it's overwritten
#endif
    }

    // Bias + leaky ReLU, write H to LDS (doubles as the layer-2 transpose).
#pragma unroll
    for (int t = 0; t < 4; ++t) {
#pragma unroll
        for (int v = 0; v < 8; ++v) {
            float h = acc[t][v] + bias1[t];
            h = fmaxf(h, 0.1f * h);              // leaky_relu, branch-free
            Hs[(wr0 + v + hi8) * HS_STRIDE + t * 16 + rl] = h;
        }
    }
    __syncthreads();

    // ---------------- Layer 2: out = leaky(H @ W2 + b2), K = 64 ----------------
    v8f o[2] = {};
#pragma unroll
    for (int kk = 0; kk < HIDDEN; kk += 4) {
        v2f a;
        const float* ap = &Hs[(wr0 + rl) * HS_STRIDE + kk + kp];
        a.x = ap[0];
        a.y = ap[1];
#pragma unroll
        for (int t = 0; t < 2; ++t) {
            v2f bb;
            bb.x = W2s[(kk + kp)     * W2S_STRIDE + t * 16 + rl];
            bb.y = W2s[(kk + kp + 1) * W2S_STRIDE + t * 16 + rl];
            o[t] = __builtin_amdgcn_wmma_f32_16x16x4_f32(
                false, a, false, bb, (short)0, o[t], false, false);
        }
    }

    // Bias + leaky + keep_count mask + store (cols >= 21 are padding, skipped).
    const int limit = min(keep_count[b], SEL);
#pragma unroll
    for (int t = 0; t < 2; ++t) {
        int c = t * 16 + rl;
        if (c < NCLS) {
            float bias2 = b2[c];
#pragma unroll
            for (int v = 0; v < 8; ++v) {
                int s = row0 + wr0 + v + hi8;
                float val = o[t][v] + bias2;
                val = fmaxf(val, 0.1f * val);
                val = (s < limit) ? val : 0.0f;  // masked rows must be exactly 0
                out[((size_t)b * SEL + s) * NCLS + c] = val;
            }
        }
    }
}

extern "C" void kernel_launch(void* const* d_in, const int* in_sizes, int n_in,
                              void* d_out, int out_size, void* d_ws, size_t ws_size,
                              hipStream_t stream) {
    const float* x          = (const float*)d_in[0];
    const float* W1         = (const float*)d_in[1];
    const float* b1         = (const float*)d_in[2];
    const float* W2         = (const float*)d_in[3];
    const float* b2         = (const float*)d_in[4];
    const int*   keep_count = (const int*)d_in[5];
    float* out = (float*)d_out;

    dim3 grid(64 * 2);   // batch x half(64 rows)
    dim3 block(128);     // 4 wave32s, one 16-row stripe each
    classifier_head_kernel<<<grid, block, 0, stream>>>(x, W1, b1, W2, b2, keep_count, out);
}